// Rotor_52072183496956
// MI455X (gfx1250) — compile-verified
//
#include <hip/hip_runtime.h>

typedef __attribute__((ext_vector_type(2)))  float v2f;
typedef __attribute__((ext_vector_type(8)))  float v8f;

#define ALPHA        26
#define S_LEN        8192
#define XSI          28     // s_in token stride: 26 rotated data + 2 zero pads (even -> b64 aligned)
#define XSO          33     // s_out token stride: cols 0..31 valid targets, gcd(33,64)=1
#define PRS          33     // s_rnB pair-row stride in v2f units
#define TOK_PER_BLK  256
#define THREADS      256

__global__ __launch_bounds__(THREADS)
void rotor_wmma_kernel(const float* __restrict__ x,
                       const float* __restrict__ rotor,
                       float* __restrict__ out)
{
    __shared__ float s_inv[ALPHA];
    __shared__ v2f   s_rnB[14 * PRS];           // pair-interleaved normalized rotor:
                                                // s_rnB[p*PRS+j] = {rn[2p][j], rn[2p+1][j]}
    __shared__ float s_in [TOK_PER_BLK * XSI];  // PRE-ROTATED input tokens + zero pads
    __shared__ float s_out[TOK_PER_BLK * XSO];  // logits (cols 26..31 dead) -> softmax

    const int  tid  = threadIdx.x;
    const int  lane = tid & 31;
    const int  wv   = tid >> 5;            // wave id 0..7, each handles 32 tokens
    const int  m    = lane & 15;           // tile row (token within 16)
    const int  hi   = lane >> 4;           // half-wave select
    const long base_tok = (long)blockIdx.x * TOK_PER_BLK;
    const int  s_base   = (int)(base_tok % S_LEN);   // blocks never straddle a batch row

    // ---- stage 0: L2-normalize rotor rows; store pair-interleaved + zero-padded ----
    if (tid < ALPHA) {
        float ss = 0.f;
        #pragma unroll
        for (int c = 0; c < ALPHA; ++c) {
            float v = rotor[tid * ALPHA + c];
            ss += v * v;
        }
        s_inv[tid] = rsqrtf(ss);
    }
    __syncthreads();
    for (int i = tid; i < 14 * PRS; i += THREADS) {
        int p = i / PRS, j = i % PRS;      // pair p covers rows 2p, 2p+1; col j (0..32)
        int r0 = 2 * p, r1 = 2 * p + 1;
        v2f v = {0.f, 0.f};
        if (j < ALPHA) {
            if (r0 < ALPHA) v.x = rotor[r0 * ALPHA + j] * s_inv[r0];
            if (r1 < ALPHA) v.y = rotor[r1 * ALPHA + j] * s_inv[r1];
        }
        s_rnB[i] = v;
    }
    // ---- stage 1: coalesced load; ROTATE while scattering into LDS ----
    // x_rot[(c + k) % 26] = x[c]  =>  WMMA A reads become contiguous.
    for (int i = tid; i < TOK_PER_BLK * ALPHA; i += THREADS) {
        int t = i / ALPHA, c = i % ALPHA;
        int k = (unsigned)(s_base + t) % ALPHA;          // rotor shift (position==1)
        int cr = c + k; if (cr >= ALPHA) cr -= ALPHA;
        s_in[t * XSI + cr] = __builtin_nontemporal_load(&x[base_tok * ALPHA + i]);
    }
    for (int t = tid; t < TOK_PER_BLK; t += THREADS) {   // K = 26,27 zero pads
        s_in[t * XSI + ALPHA]     = 0.f;
        s_in[t * XSI + ALPHA + 1] = 0.f;
    }
    __syncthreads();

    // ---- stage 2: B fragments, one ds_load_b64 each (whole-kernel invariant) ----
    // fp32 WMMA 16x16x4 layout, A and B both: K = 4c + 2*(lane/16) + vgpr
    const int j0 = m;                      // column within 16-wide N tile
    v2f Bf0[7], Bf1[7];                    // cols 0..15 and 16..31 (>=26 are zero)
    #pragma unroll
    for (int c = 0; c < 7; ++c) {
        const int p = 2 * c + hi;          // pair index = K/2
        Bf0[c] = s_rnB[p * PRS + j0];
        Bf1[c] = s_rnB[p * PRS + j0 + 16];
    }

    // ---- stage 3: two 16-token WMMA tiles per wave ----
    #pragma unroll
    for (int t2 = 0; t2 < 2; ++t2) {
        const int tokL = wv * 32 + t2 * 16 + m;          // token within block
        // pre-rotated row, 8-byte aligned pairs: a = x_rot[K], x_rot[K+1]
        const v2f* aw = (const v2f*)(s_in + tokL * XSI);

        v8f acc0 = {};                     // cols 0..15
        v8f acc1 = {};                     // cols 16..25 (26..31 zero-padded)
        #pragma unroll
        for (int c = 0; c < 7; ++c) {
            v2f a = aw[2 * c + hi];        // K = 4c + 2*hi, single ds_load_b64
            acc0 = __builtin_amdgcn_wmma_f32_16x16x4_f32(
                       false, a, false, Bf0[c], (short)0, acc0, false, false);
            acc1 = __builtin_amdgcn_wmma_f32_16x16x4_f32(
                       false, a, false, Bf1[c], (short)0, acc1, false, false);
        }

        // D tile -> LDS, unconditional (cols 26..31 are dead scratch)
        // C/D layout: vgpr v = row v (lo) / v+8 (hi), col = lane&15
        #pragma unroll
        for (int v = 0; v < 8; ++v) {
            const int row = wv * 32 + t2 * 16 + v + 8 * hi;
            s_out[row * XSO + j0]      = acc0[v];
            s_out[row * XSO + j0 + 16] = acc1[v];
        }
    }
    __syncthreads();

    // ---- stage 4: softmax over 26 logits, one thread per token ----
    {
        float vals[ALPHA];
        float mx = -3.4e38f;
        #pragma unroll
        for (int j = 0; j < ALPHA; ++j) {
            vals[j] = s_out[tid * XSO + j];
            mx = fmaxf(mx, vals[j]);
        }
        float sum = 0.f;
        #pragma unroll
        for (int j = 0; j < ALPHA; ++j) {
            vals[j] = __expf(vals[j] - mx);
            sum += vals[j];
        }
        const float inv = 1.f / sum;
        #pragma unroll
        for (int j = 0; j < ALPHA; ++j)
            s_out[tid * XSO + j] = vals[j] * inv;
    }
    __syncthreads();

    // ---- stage 5: coalesced non-temporal store ----
    for (int i = tid; i < TOK_PER_BLK * ALPHA; i += THREADS) {
        int t = i / ALPHA, c = i % ALPHA;
        __builtin_nontemporal_store(s_out[t * XSO + c], &out[base_tok * ALPHA + i]);
    }
}

extern "C" void kernel_launch(void* const* d_in, const int* in_sizes, int n_in,
                              void* d_out, int out_size, void* d_ws, size_t ws_size,
                              hipStream_t stream) {
    (void)n_in; (void)out_size; (void)d_ws; (void)ws_size;
    const float* x     = (const float*)d_in[0];   // [B,S,26] fp32
    const float* rotor = (const float*)d_in[1];   // [26,26] fp32
    float* out = (float*)d_out;                   // [B,S,26] fp32

    const int total_tokens = in_sizes[0] / ALPHA;      // B*S = 1,048,576
    const int blocks = total_tokens / TOK_PER_BLK;     // 4096
    rotor_wmma_kernel<<<blocks, THREADS, 0, stream>>>(x, rotor, out);
}